// AspectAwareAttention_22746146799925
// MI455X (gfx1250) — compile-verified
//
#include <hip/hip_runtime.h>
#include <math.h>

typedef __attribute__((ext_vector_type(2))) float v2f;
typedef __attribute__((ext_vector_type(8))) float v8f;

constexpr int cB = 8, cT = 2048, cH = 768, cA = 8, cNH = 4, cHD = 192;
constexpr int cR = cA * cNH;      // 32 fused (aspect,head) rows
constexpr int c3H = 3 * cH;       // 2304

// workspace offsets (in floats)
constexpr size_t OFF_MQ    = 0;                            // cR*cH        = 24576
constexpr size_t OFF_QB    = OFF_MQ + (size_t)cR * cH;     // cR           = 32
constexpr size_t OFF_ATTN  = OFF_QB + 128;                 // cB*cR*cT     = 524288
constexpr size_t OFF_SPOOL = OFF_ATTN + (size_t)cB*cR*cT;  // cB*cR*cH     = 196608
constexpr size_t OFF_CTX   = OFF_SPOOL + (size_t)cB*cR*cH; // cB*cA*cH     = 49152

__device__ __forceinline__ v8f wmma_f32x4(v2f a, v2f b, v8f c) {
  // V_WMMA_F32_16X16X4_F32 : D = A(16x4,f32) * B(4x16,f32) + C(16x16,f32)
  return __builtin_amdgcn_wmma_f32_16x16x4_f32(false, a, false, b, (short)0, c,
                                               false, false);
}

// CDNA5 async global->LDS DMA (GLOBAL_LOAD_ASYNC_TO_LDS_B128, ASYNCcnt-tracked).
// saddr form: per-lane 32-bit byte offset VGPR + uniform 64-bit SGPR base.
// LDS destination = per-lane VGPR holding the LDS byte address (addr[31:0]).
__device__ __forceinline__ void async_ld_b128(void* lds_dst, const void* gbase,
                                              unsigned voff_bytes) {
  unsigned lds = (unsigned)(uintptr_t)lds_dst;
  asm volatile("global_load_async_to_lds_b128 %0, %1, %2"
               :: "v"(lds), "v"(voff_bytes), "s"(gbase)
               : "memory");
}
__device__ __forceinline__ void wait_async0() {
  asm volatile("s_wait_asynccnt 0x0" ::: "memory");
}

// ---------------------------------------------------------------------------
// k1: q[a,e] = aq[a]·Wq[a,e,:] + bq, scaled by 1/sqrt(hd);
//     mq[a*4+n, h] = sum_d q[a,n,d] * Wk[a, n*hd+d, h]   (q-folded key proj)
//     qb[a*4+n]    = sum_d q[a,n,d] * bk[a, n*hd+d]
__global__ void k_qprep(const float* __restrict__ aq,
                        const float* __restrict__ Wqkv,
                        const float* __restrict__ bqkv,
                        float* __restrict__ ws) {
  const int a = blockIdx.x, tid = threadIdx.x;
  __shared__ float qs[cH];
  const float scale = 0.07216878364870322f;  // 1/sqrt(192)
  for (int i = 0; i < 3; ++i) {
    int e = tid + i * 256;
    float acc = 0.f;
    const float* wrow = Wqkv + (size_t)(a * c3H + e) * cH;
    for (int h = 0; h < cH; ++h) acc += aq[a * cH + h] * wrow[h];
    qs[e] = (acc + bqkv[a * c3H + e]) * scale;
  }
  __syncthreads();
  for (int i = 0; i < 12; ++i) {
    int idx = tid + i * 256;              // 0..3071 = n*768 + h
    int n = idx / cH, h = idx - n * cH;
    float acc = 0.f;
    for (int d = 0; d < cHD; ++d)
      acc += qs[n * cHD + d] * Wqkv[(size_t)(a * c3H + cH + n * cHD + d) * cH + h];
    ws[OFF_MQ + (size_t)(a * cNH + n) * cH + h] = acc;
  }
  if (tid < cNH) {
    int n = tid;
    float acc = 0.f;
    for (int d = 0; d < cHD; ++d)
      acc += qs[n * cHD + d] * bqkv[a * c3H + cH + n * cHD + d];
    ws[OFF_QB + a * cNH + n] = acc;
  }
}

// ---------------------------------------------------------------------------
// k2: scores[b, r, t] = mq[r,:]·seq[b,t,:] + qb[r], masked.
// grid (16, B, 2): x = 128-token group, y = batch, z = which 16 of the 32 rows.
// 8 waves/WG, each wave owns a 16-token tile; 1 fp32 WMMA per K-step of 4.
constexpr int MQS = 772;  // pad: 772%64==4 -> conflict-free b64 A-frag reads,
                          // and 772%4==0 -> 16B-aligned B128 async staging.
__global__ void k_scores(const float* __restrict__ seq,
                         const int* __restrict__ amask,
                         float* __restrict__ ws) {
  __shared__ float mqs[16 * MQS];
  const int tid = threadIdx.x, b = blockIdx.y, rbase = blockIdx.z * 16;
  // async DMA the 16x768 mq slab straight into LDS (no VGPR staging)
  const float* gmq = ws + OFF_MQ + (size_t)rbase * cH;
  for (int i = 0; i < 12; ++i) {
    int idx = (tid + i * 256) * 4;        // covers 12288 floats, 4 per lane
    int r = idx / cH, h = idx - r * cH;   // rows are 768 = 192 chunks of 4
    async_ld_b128(&mqs[r * MQS + h], gmq, (unsigned)idx * 4u);
  }
  wait_async0();
  __syncthreads();

  const int wave = tid >> 5, lane = tid & 31;
  const int lr = lane & 15, lh = lane >> 4;
  const int t = (blockIdx.x * 8 + wave) * 16 + lr;
  const float* bp = seq + ((size_t)b * cT + t) * cH + 2 * lh;  // B-frag: seq^T
  const float* ap = mqs + lr * MQS + 2 * lh;                   // A-frag: mq rows
  v8f c = {};
#pragma unroll 4
  for (int k = 0; k < cH / 4; ++k) {
    v2f av = *(const v2f*)(ap + 4 * k);
    v2f bv = *(const v2f*)(bp + 4 * k);
    c = wmma_f32x4(av, bv, c);
  }
  // branch-free masked epilogue: hoist qb loads, then pure selects
  float qbv[8];
#pragma unroll
  for (int j = 0; j < 8; ++j) qbv[j] = ws[OFF_QB + rbase + 8 * lh + j];
  const int m = amask[b * cT + t];
  float* orow = ws + OFF_ATTN + (size_t)b * cR * cT + t;
#pragma unroll
  for (int j = 0; j < 8; ++j) {
    int r = rbase + 8 * lh + j;           // C layout: lane col = t, row = 8*lh+j
    float v = c[j] + qbv[j];
    if (m == 0) v = -1.0e9f;
    orow[(size_t)r * cT] = v;
  }
}

// ---------------------------------------------------------------------------
// k3: softmax over t for each (b,a,n) row (in-place), plus head-mean attn out.
__device__ __forceinline__ float blk_red256(float v, float* red, bool domax) {
  int tid = threadIdx.x;
  red[tid] = v; __syncthreads();
  for (int s = 128; s > 0; s >>= 1) {
    if (tid < s) red[tid] = domax ? fmaxf(red[tid], red[tid + s])
                                  : (red[tid] + red[tid + s]);
    __syncthreads();
  }
  float r = red[0]; __syncthreads();
  return r;
}

__global__ void k_softmax(float* __restrict__ ws, float* __restrict__ attnw) {
  __shared__ float red[256];
  const int a = blockIdx.x, b = blockIdx.y, tid = threadIdx.x;
  float macc[8];
#pragma unroll
  for (int i = 0; i < 8; ++i) macc[i] = 0.f;
  for (int n = 0; n < cNH; ++n) {
    float* row = ws + OFF_ATTN + (size_t)(b * cR + a * cNH + n) * cT;
    float v[8], mx = -3.4e38f;
#pragma unroll
    for (int i = 0; i < 8; ++i) { v[i] = row[tid + i * 256]; mx = fmaxf(mx, v[i]); }
    mx = blk_red256(mx, red, true);
    float ls = 0.f;
#pragma unroll
    for (int i = 0; i < 8; ++i) { v[i] = __expf(v[i] - mx); ls += v[i]; }
    ls = blk_red256(ls, red, false);
    float inv = 1.f / ls;
#pragma unroll
    for (int i = 0; i < 8; ++i) {
      float av = v[i] * inv;
      row[tid + i * 256] = av;
      macc[i] += av * 0.25f;
    }
  }
  float* orow = attnw + (size_t)(a * cB + b) * cT;
#pragma unroll
  for (int i = 0; i < 8; ++i) orow[tid + i * 256] = macc[i];
}

// ---------------------------------------------------------------------------
// k4: s_pool[b, r, h] = sum_t attn[b,r,t] * seq[b,t,h]
// grid (6, B); 8 waves/WG, wave owns a 16-wide h tile; 2 M-tiles (r 0-15,16-31)
// per wave => 2 fp32 WMMAs per K-step of 4 tokens, K-loop = 512.
__global__ void k_pool(const float* __restrict__ seq, float* __restrict__ ws) {
  const int tid = threadIdx.x, b = blockIdx.y;
  const int wave = tid >> 5, lane = tid & 31;
  const int lr = lane & 15, lh = lane >> 4;
  const int h0 = (blockIdx.x * 8 + wave) * 16;
  const float* a0p = ws + OFF_ATTN + (size_t)(b * cR + lr) * cT + 2 * lh;
  const float* a1p = a0p + (size_t)16 * cT;
  const float* bp  = seq + ((size_t)b * cT + 2 * lh) * cH + h0 + lr;
  v8f c0 = {}, c1 = {};
  for (int k = 0; k < cT / 4; ++k) {
    v2f av0 = *(const v2f*)(a0p + 4 * k);
    v2f av1 = *(const v2f*)(a1p + 4 * k);
    v2f bv;
    bv.x = bp[(size_t)k * 4 * cH];        // token 4k+2*lh
    bv.y = bp[(size_t)k * 4 * cH + cH];   // token 4k+2*lh+1
    c0 = wmma_f32x4(av0, bv, c0);
    c1 = wmma_f32x4(av1, bv, c1);
  }
  float* op = ws + OFF_SPOOL + (size_t)b * cR * cH + h0 + lr;
#pragma unroll
  for (int j = 0; j < 8; ++j) {
    op[(size_t)(8 * lh + j) * cH]      = c0[j];
    op[(size_t)(16 + 8 * lh + j) * cH] = c1[j];
  }
}

// ---------------------------------------------------------------------------
// k5a: ctx[b,a,e] = Wv[a,e,:]·s_pool[b,a,head(e),:] + bv[a,e]
// grid (A, 2): each WG serves 4 batches, reusing each weight element 4x.
__global__ void k_ctx(const float* __restrict__ Wqkv,
                      const float* __restrict__ bqkv,
                      float* __restrict__ ws) {
  __shared__ float sl[4 * cNH * cH];      // 48 KB
  const int a = blockIdx.x, bbase = blockIdx.y * 4, tid = threadIdx.x;
  for (int i = 0; i < 4; ++i) {
    int idx = (tid + i * 768) * 4;        // 0..12284, 4 floats per lane
    int bl = idx / (cNH * cH);
    int rem = idx - bl * (cNH * cH);
    int n = rem / cH, h = rem - n * cH;   // h multiple of 4, rows divisible
    unsigned goff = (unsigned)((((bbase + bl) * cR + a * cNH + n) * cH + h) * 4);
    async_ld_b128(&sl[idx], ws + OFF_SPOOL, goff);
  }
  wait_async0();
  __syncthreads();
  const int e = tid, ne = e / cHD;
  const int row = a * c3H + 2 * cH + e;   // Wv row
  const float bv = bqkv[row];
  float acc[4] = {bv, bv, bv, bv};
  const float* wr = Wqkv + (size_t)row * cH;
  for (int h = 0; h < cH; ++h) {
    float w = wr[h];
#pragma unroll
    for (int j = 0; j < 4; ++j) acc[j] += w * sl[(j * cNH + ne) * cH + h];
  }
#pragma unroll
  for (int j = 0; j < 4; ++j)
    ws[OFF_CTX + (size_t)((bbase + j) * cA + a) * cH + e] = acc[j];
}

// ---------------------------------------------------------------------------
// k5b: out[b,a,e] = Wo[a,e,:]·ctx[b,a,:] + bo, then LayerNorm, store (A,B,H).
__device__ __forceinline__ float blk_sum768(float v, float* red) {
  int tid = threadIdx.x;
  red[tid] = v;
  if (tid < 256) red[768 + tid] = 0.f;
  __syncthreads();
  for (int s = 512; s > 0; s >>= 1) {
    if (tid < s) red[tid] += red[tid + s];
    __syncthreads();
  }
  float r = red[0]; __syncthreads();
  return r;
}

__global__ void k_out_ln(const float* __restrict__ Wo,
                         const float* __restrict__ bo,
                         const float* __restrict__ gamma,
                         const float* __restrict__ beta,
                         float* __restrict__ ws,
                         float* __restrict__ out) {
  __shared__ float ctxs[cB * cH];         // 24 KB
  __shared__ float red[1024];
  const int a = blockIdx.x, tid = threadIdx.x;
  for (int i = 0; i < 2; ++i) {
    int idx = (tid + i * 768) * 4;        // 0..6140, 4 floats per lane
    int bl = idx / cH, e = idx - bl * cH;
    unsigned goff = (unsigned)(((bl * cA + a) * cH + e) * 4);
    async_ld_b128(&ctxs[idx], ws + OFF_CTX, goff);
  }
  wait_async0();
  __syncthreads();
  const int e = tid;
  const size_t row = (size_t)(a * cH + e);
  const float bov = bo[row];
  float acc[cB];
#pragma unroll
  for (int j = 0; j < cB; ++j) acc[j] = bov;
  const float* wr = Wo + row * cH;
  for (int h = 0; h < cH; ++h) {
    float w = wr[h];
#pragma unroll
    for (int j = 0; j < cB; ++j) acc[j] += w * ctxs[j * cH + h];
  }
  const float g = gamma[e], bt = beta[e];
  for (int j = 0; j < cB; ++j) {
    float mu = blk_sum768(acc[j], red) * (1.f / 768.f);
    float d = acc[j] - mu;
    float var = blk_sum768(d * d, red) * (1.f / 768.f);
    out[(size_t)(a * cB + j) * cH + e] = d * rsqrtf(var + 1e-5f) * g + bt;
  }
}

// ---------------------------------------------------------------------------
extern "C" void kernel_launch(void* const* d_in, const int* in_sizes, int n_in,
                              void* d_out, int out_size, void* d_ws, size_t ws_size,
                              hipStream_t stream) {
  const float* seq   = (const float*)d_in[0];
  const int*   amask = (const int*)d_in[1];   // attention_mask (integer)
  const float* aq    = (const float*)d_in[2];
  const float* Wqkv  = (const float*)d_in[3];
  const float* bqkv  = (const float*)d_in[4];
  const float* Wo    = (const float*)d_in[5];
  const float* bo    = (const float*)d_in[6];
  const float* gamma = (const float*)d_in[7];
  const float* beta  = (const float*)d_in[8];
  float* out = (float*)d_out;                        // [A,B,H] then [A,B,T]
  float* ws  = (float*)d_ws;                         // ~3.2 MB used
  float* attnw = out + (size_t)cA * cB * cH;

  k_qprep  <<<dim3(cA),        dim3(256), 0, stream>>>(aq, Wqkv, bqkv, ws);
  k_scores <<<dim3(16, cB, 2), dim3(256), 0, stream>>>(seq, amask, ws);
  k_softmax<<<dim3(cA, cB),    dim3(256), 0, stream>>>(ws, attnw);
  k_pool   <<<dim3(6, cB),     dim3(256), 0, stream>>>(seq, ws);
  k_ctx    <<<dim3(cA, 2),     dim3(768), 0, stream>>>(Wqkv, bqkv, ws);
  k_out_ln <<<dim3(cA),        dim3(768), 0, stream>>>(Wo, bo, gamma, beta, ws, out);
}